// DCT2_6579889897985
// MI455X (gfx1250) — compile-verified
//
#include <hip/hip_runtime.h>
#include <hip/hip_bf16.h>

typedef float v2f __attribute__((ext_vector_type(2)));
typedef float v4f __attribute__((ext_vector_type(4)));
typedef float v8f __attribute__((ext_vector_type(8)));

#define IMG_W 512
#define IMG_H 512
#define CHW   (IMG_W * IMG_H)

// One wave32 handles one 16x16 tile (= four 8x8 DCT blocks) for all 3 channels.
//   Y   = M[oc] . X + off[oc]          (BT.601, folded into B-operand build, FMA chain)
//   T   = Q2 * Y                       (WMMA f32 16x16x4, K looped 4x)
//   U   = Q2 * T^T                     (T^T staged via wave-private padded LDS)
//   out = U^T                          (two b128 NT stores per lane per channel)
// Q2 = blockdiag(Q8, Q8) so the 16x16 product == per-8x8-block Q*blk*Q^T.
// No workgroup barriers: LDS patches are wave-private and same-wave DS ops are
// in-order at the LDS unit, so an LDS-only fence (compiler ordering) suffices.
__global__ __launch_bounds__(256) void dct8_ycbcr_wmma(const float* __restrict__ x,
                                                       float* __restrict__ out,
                                                       int tiles_total) {
    const int lane = threadIdx.x & 31;
    const int h    = lane >> 4;    // lane half (0/1)
    const int n    = lane & 15;    // N column (B/C/D) and M row (A)
    const int wave = threadIdx.x >> 5;

    const int t = blockIdx.x * 8 + wave;           // tile id (grid sized exactly)
    const int bimg = t >> 10;                      // 1024 tiles per image (32x32)
    const int rem  = t & 1023;
    const int ty   = rem >> 5;
    const int tx   = rem & 31;

    const float* xb = x   + (size_t)bimg * (3 * CHW) + (size_t)(ty * 16) * IMG_W + tx * 16;
    float*       ob = out + (size_t)bimg * (3 * CHW) + (size_t)(ty * 16 + n) * IMG_W + tx * 16 + 8 * h;

    // ---- Q2 in WMMA A-layout: qa[2*kc+v] = Q2[m=n][4*kc + 2*h + v] ----
    float qa[8];
#pragma unroll
    for (int p = 0; p < 8; ++p) {
        const int col = 4 * (p >> 1) + 2 * h + (p & 1);
        const int m   = n;
        float q = 0.0f;
        if ((m < 8) == (col < 8)) {                 // block-diagonal
            const int i = m & 7, j = col & 7;
            q = (i == 0) ? 0.35355339059327379f     // sqrt(1/8)
                         : 0.5f * __cosf((float)((2 * j + 1) * i) * 0.19634954084936207f); // pi/16
        }
        qa[p] = q;
    }

    // ---- load 3 channels in WMMA B-layout: xv[c][2*kc+v] = X_c[4*kc+2*h+v][n] ----
    float xv[3][8];
#pragma unroll
    for (int c = 0; c < 3; ++c) {
#pragma unroll
        for (int p = 0; p < 8; ++p) {
            const int row = 4 * (p >> 1) + 2 * h + (p & 1);
            xv[c][p] = __builtin_nontemporal_load(xb + (size_t)c * CHW + row * IMG_W + n);
        }
    }

    // wave-private, per-channel LDS patches: no WAR hazard across channels
    __shared__ float lds[8 * 3 * 272];              // 16x17 padded patch each
    float* Lw = lds + wave * (3 * 272);

    const float M[3][3]  = {{ 0.299f,     0.587f,     0.114f    },
                            {-0.168736f, -0.331264f,  0.5f      },
                            { 0.5f,      -0.418688f, -0.081312f }};
    const float OFF[3]   = { 0.0f, 128.0f / 255.0f, 128.0f / 255.0f };

#pragma unroll
    for (int oc = 0; oc < 3; ++oc) {
        float* L = Lw + oc * 272;

        // ---- T = Q2 * Y : K=16 in four K=4 WMMA steps ----
        v8f acc = {0.f, 0.f, 0.f, 0.f, 0.f, 0.f, 0.f, 0.f};
#pragma unroll
        for (int kc = 0; kc < 4; ++kc) {
            v2f a;  a[0] = qa[2 * kc];  a[1] = qa[2 * kc + 1];
            v2f bv;
            bv[0] = fmaf(M[oc][0], xv[0][2 * kc],
                    fmaf(M[oc][1], xv[1][2 * kc],
                    fmaf(M[oc][2], xv[2][2 * kc], OFF[oc])));
            bv[1] = fmaf(M[oc][0], xv[0][2 * kc + 1],
                    fmaf(M[oc][1], xv[1][2 * kc + 1],
                    fmaf(M[oc][2], xv[2][2 * kc + 1], OFF[oc])));
            acc = __builtin_amdgcn_wmma_f32_16x16x4_f32(false, a, false, bv, (short)0, acc, false, false);
        }

        // ---- stage T (C-layout: acc[v] = T[v+8h][n]) row-major, stride 17 ----
#pragma unroll
        for (int v = 0; v < 8; ++v) L[(v + 8 * h) * 17 + n] = acc[v];

        // LDS-only fence: compiler ordering for the cross-lane store->load handoff;
        // same-wave DS ops are in-order at the LDS unit (no workgroup barrier needed).
        __builtin_amdgcn_fence(__ATOMIC_ACQ_REL, "workgroup", "local");

        // ---- U = Q2 * T^T : b[v] = T[n][4*kc + 2*h + v] ----
        v8f acc2 = {0.f, 0.f, 0.f, 0.f, 0.f, 0.f, 0.f, 0.f};
#pragma unroll
        for (int kc = 0; kc < 4; ++kc) {
            v2f a;  a[0] = qa[2 * kc];  a[1] = qa[2 * kc + 1];
            const int c0 = 4 * kc + 2 * h;
            v2f bv; bv[0] = L[n * 17 + c0]; bv[1] = L[n * 17 + c0 + 1];
            acc2 = __builtin_amdgcn_wmma_f32_16x16x4_f32(false, a, false, bv, (short)0, acc2, false, false);
        }

        // ---- out = U^T : acc2[v] = U[v+8h][n] = out[n][v+8h] -> 8 contiguous floats ----
        v4f s0; s0[0] = acc2[0]; s0[1] = acc2[1]; s0[2] = acc2[2]; s0[3] = acc2[3];
        v4f s1; s1[0] = acc2[4]; s1[1] = acc2[5]; s1[2] = acc2[6]; s1[3] = acc2[7];
        float* op = ob + (size_t)oc * CHW;
        __builtin_nontemporal_store(s0, (v4f*)op);
        __builtin_nontemporal_store(s1, (v4f*)(op + 4));
    }
    (void)tiles_total;
}

extern "C" void kernel_launch(void* const* d_in, const int* in_sizes, int n_in,
                              void* d_out, int out_size, void* d_ws, size_t ws_size,
                              hipStream_t stream) {
    const float* x = (const float*)d_in[0];
    float* out = (float*)d_out;

    // (B,3,512,512): tiles = B * (512/16)^2, one wave per tile, 8 waves per block
    const int batches = in_sizes[0] / (3 * CHW);
    const int tiles   = batches * (IMG_H / 16) * (IMG_W / 16);
    const int blocks  = tiles / 8;   // 64*1024/8 = 8192 for the reference shape

    dct8_ycbcr_wmma<<<blocks, 256, 0, stream>>>(x, out, tiles);

    (void)n_in; (void)out_size; (void)d_ws; (void)ws_size;
}